// EWACFLoss_60103772340524
// MI455X (gfx1250) — compile-verified
//
#include <hip/hip_runtime.h>
#include <hip/hip_bf16.h>

#define DEVFN __device__ __forceinline__

namespace {
constexpr int B = 64, T = 4096, F = 64, H = 256;
constexpr int CHUNK = 256, WARM = 64, NCHUNK = T / CHUNK;
constexpr float LAM   = 0.5f;
constexpr float EPSC  = 1e-8f;
constexpr float ALPHA = 0.5f;
constexpr int NB_PEN = 4096, NT_PEN = 256;   // 4096 blocks x 256 thr stream 256 MB
constexpr int NB_MSE = 256,  NT_MSE = 256;
} // namespace

// ---------------- CDNA5 feature detection ----------------------------------
#if defined(__has_builtin)
#  if __has_builtin(__builtin_amdgcn_global_load_async_to_lds_b128)
#    define HAVE_ASYNC_BUILTIN 1
#  endif
#  if __has_builtin(__builtin_amdgcn_s_wait_asynccnt)
#    define HAVE_WAITASYNC_BUILTIN 1
#  endif
#  if __has_builtin(__builtin_amdgcn_wmma_f32_16x16x4_f32)
#    define HAVE_WMMA4 1
#  endif
#endif

typedef int   v4i __attribute__((ext_vector_type(4)));
typedef float v2f __attribute__((ext_vector_type(2)));
typedef float v8f __attribute__((ext_vector_type(8)));
typedef __attribute__((address_space(1))) v4i as1_v4i;
typedef __attribute__((address_space(3))) v4i as3_v4i;

DEVFN as1_v4i* to_as1_v4i(const void* p) {
  return (as1_v4i*)(unsigned long long)p;            // generic global VA == AS1 VA
}
DEVFN as3_v4i* to_as3_v4i(const void* p) {
  // generic LDS address = {SHARED_BASE hi32, lds_offset}; low 32 bits = LDS byte addr
  return (as3_v4i*)(unsigned int)(unsigned long long)p;
}

DEVFN void async_copy_b128(const void* gsrc, void* ldst) {
#if defined(HAVE_ASYNC_BUILTIN)
  __builtin_amdgcn_global_load_async_to_lds_b128(to_as1_v4i(gsrc), to_as3_v4i(ldst), 0, 0);
#else
  unsigned           laddr = (unsigned)(unsigned long long)ldst;
  unsigned long long gaddr = (unsigned long long)gsrc;
  asm volatile("global_load_async_to_lds_b128 %0, %1, off"
               :: "v"(laddr), "v"(gaddr) : "memory");
#endif
}

template <int N> DEVFN void wait_asynccnt() {
#if defined(HAVE_WAITASYNC_BUILTIN)
  __builtin_amdgcn_s_wait_asynccnt((unsigned short)N);
  asm volatile("" ::: "memory");
#else
  asm volatile("s_wait_asynccnt %0" :: "n"(N) : "memory");
#endif
}

// ---------------- Kernel 1: chunked EWMA autocorr scan ---------------------
// grid = B*NCHUNK blocks of F threads; thread f scans feature f of batch b.
// lam=0.5 => 64-step warm-up makes chunk-seam error ~2^-64 (<< fp32 eps).
// F-mean is done with V_WMMA_F32_16X16X4_F32: rowsum = A(16x4 chunks) x ones.
__global__ __launch_bounds__(F) void ewacf_scan_kernel(
    const float* __restrict__ seq, float* __restrict__ irr) {
  __shared__ float tile[CHUNK][F + 1];               // +1 pad: conflict-free access
  const int f     = threadIdx.x;
  const int b     = blockIdx.x / NCHUNK;
  const int chunk = blockIdx.x % NCHUNK;
  const int t0    = chunk * CHUNK;
  const int ts    = (chunk == 0) ? 1 : (t0 - WARM);

  const float* base = seq + (size_t)b * T * F + f;

  if (chunk == 0) tile[0][f] = 1.0f;                 // pad row: ac=0 -> irr=1

  float m = 0.0f, v = EPSC, c = 0.0f;
  float xl = base[(size_t)(ts - 1) * F];
  const int te = t0 + CHUNK;
  for (int t = ts; t < te; ++t) {
    if (((t & 7) == 0) && (t + 16 < T))              // gfx1250 global_prefetch_b8
      __builtin_prefetch(base + (size_t)(t + 16) * F, 0, 1);
    const float xt = base[(size_t)t * F];
    m = LAM * m + (1.0f - LAM) * xt;
    const float d = xt - m;
    v = LAM * v + (1.0f - LAM) * d * d;
    c = LAM * c + (1.0f - LAM) * d * (xl - m);
    if (t >= t0) {
      const float ac = c / sqrtf(v * v + EPSC);
      tile[t - t0][f] = 1.0f - fabsf(ac);
    }
    xl = xt;
  }
  __syncthreads();

#if defined(HAVE_WMMA4)
  // Mean over F via WMMA. Per wave: 8 tiles of 16 rows. For each tile,
  // chain 16 accumulating 16x16x4 WMMAs with B = ones => D[m,n] = rowsum(m).
  // A layout (32-bit 16x4): lanes 0-15 -> K={0,1}, lanes 16-31 -> K={2,3}, M=lane&15.
  // D layout: lane<16 -> c[j]=D[M=j][N=lane]; lane>=16 -> c[j]=D[M=8+j][.].
  const int wid  = threadIdx.x >> 5;
  const int lane = threadIdx.x & 31;
  const int mrow = lane & 15;
  const int koff = (lane >> 4) << 1;                 // 0 or 2
  const v2f bones = {1.0f, 1.0f};
  for (int q = 0; q < 8; ++q) {
    const int r0 = (wid * 8 + q) * 16;
    v8f acc = {0.f, 0.f, 0.f, 0.f, 0.f, 0.f, 0.f, 0.f};
#pragma unroll
    for (int kc = 0; kc < 16; ++kc) {
      const int col = 4 * kc + koff;
      v2f a;
      a[0] = tile[r0 + mrow][col];
      a[1] = tile[r0 + mrow][col + 1];
      acc = __builtin_amdgcn_wmma_f32_16x16x4_f32(
          /*neg_a=*/false, a, /*neg_b=*/false, bones,
          /*c_mod=*/(short)0, acc, /*reuse_a=*/false, /*reuse_b=*/false);
    }
    if ((lane & 15) == 0) {                          // lane 0 -> rows r0..r0+7, lane 16 -> +8..+15
      float* dst = irr + (size_t)b * T + t0 + r0 + ((lane >> 4) << 3);
      float4 lo, hi;
      lo.x = acc[0] * (1.0f / F); lo.y = acc[1] * (1.0f / F);
      lo.z = acc[2] * (1.0f / F); lo.w = acc[3] * (1.0f / F);
      hi.x = acc[4] * (1.0f / F); hi.y = acc[5] * (1.0f / F);
      hi.z = acc[6] * (1.0f / F); hi.w = acc[7] * (1.0f / F);
      ((float4*)dst)[0] = lo;
      ((float4*)dst)[1] = hi;
    }
  }
#else
  for (int r = f; r < CHUNK; r += F) {               // scalar fallback
    float s = 0.0f;
    for (int j = 0; j < F; ++j) s += tile[r][j];
    irr[(size_t)b * T + t0 + r] = s * (1.0f / F);
  }
#endif
}

// ---------------- Kernel 2: stream forget_gates via async->LDS -------------
// Each lane double-buffers its own 16B staging slot; ASYNCcnt paces the pipe.
__global__ __launch_bounds__(NT_PEN) void ewacf_penalty_kernel(
    const float* __restrict__ fg, const float* __restrict__ irr,
    float* __restrict__ partials) {
  __shared__ float4 stage[2][NT_PEN];                // 8 KB bounce buffers
  __shared__ float  wred[NT_PEN / 32];

  const int tid = threadIdx.x;
  const long long nvec   = (long long)B * T * H / 4; // float4 elements
  const long long stride = (long long)gridDim.x * blockDim.x;
  long long i = (long long)blockIdx.x * blockDim.x + tid;

  float acc = 0.0f;
  if (i < nvec) async_copy_b128(fg + 4 * i, &stage[0][tid]);
  int buf = 0;
  for (; i < nvec; i += stride) {
    const long long inext = i + stride;
    if (inext < nvec) {                              // wave-uniform (sizes divide)
      async_copy_b128(fg + 4 * inext, &stage[buf ^ 1][tid]);
      wait_asynccnt<1>();                            // current tile landed
    } else {
      wait_asynccnt<0>();                            // drain for last tile
    }
    const float4 xv = stage[buf][tid];
    const float  w  = irr[i >> 6];                   // (4i)/H, H=256
    acc += w * ((xv.x + xv.y) + (xv.z + xv.w));
    buf ^= 1;
  }
  wait_asynccnt<0>();

  for (int off = 16; off > 0; off >>= 1) acc += __shfl_xor(acc, off, 32);
  if ((tid & 31) == 0) wred[tid >> 5] = acc;
  __syncthreads();
  if (tid == 0) {
    float s = 0.0f;
    for (int k = 0; k < NT_PEN / 32; ++k) s += wred[k];
    partials[blockIdx.x] = s;                        // fixed slot: deterministic
  }
}

// ---------------- Kernel 3: MSE partials -----------------------------------
__global__ __launch_bounds__(NT_MSE) void ewacf_mse_kernel(
    const float* __restrict__ a, const float* __restrict__ b,
    float* __restrict__ partials) {
  __shared__ float wred[NT_MSE / 32];
  const int tid = threadIdx.x;
  const long long nvec   = (long long)B * T / 4;
  const long long stride = (long long)gridDim.x * blockDim.x;
  const float4* a4 = (const float4*)a;
  const float4* b4 = (const float4*)b;
  float acc = 0.0f;
  for (long long i = (long long)blockIdx.x * blockDim.x + tid; i < nvec; i += stride) {
    const float4 x = a4[i], y = b4[i];
    const float dx = x.x - y.x, dy = x.y - y.y, dz = x.z - y.z, dw = x.w - y.w;
    acc += (dx * dx + dy * dy) + (dz * dz + dw * dw);
  }
  for (int off = 16; off > 0; off >>= 1) acc += __shfl_xor(acc, off, 32);
  if ((tid & 31) == 0) wred[tid >> 5] = acc;
  __syncthreads();
  if (tid == 0) {
    float s = 0.0f;
    for (int k = 0; k < NT_MSE / 32; ++k) s += wred[k];
    partials[blockIdx.x] = s;
  }
}

// ---------------- Kernel 4: deterministic finalize --------------------------
__global__ __launch_bounds__(256) void ewacf_finalize_kernel(
    const float* __restrict__ pP, const float* __restrict__ pM,
    float* __restrict__ out) {
  __shared__ double sps[256], sms[256];
  const int t = threadIdx.x;
  double sp = 0.0, sm = 0.0;
  for (int i = t; i < NB_PEN; i += 256) sp += (double)pP[i];
  for (int i = t; i < NB_MSE; i += 256) sm += (double)pM[i];
  sps[t] = sp; sms[t] = sm;
  __syncthreads();
  if (t == 0) {
    double SP = 0.0, SM = 0.0;
    for (int i = 0; i < 256; ++i) { SP += sps[i]; SM += sms[i]; }
    const double mse = SM / (double)((long long)B * T);
    const double pen = SP / (double)((long long)B * T * H);
    out[0] = (float)(mse + (double)ALPHA * pen);
  }
}

extern "C" void kernel_launch(void* const* d_in, const int* in_sizes, int n_in,
                              void* d_out, int out_size, void* d_ws, size_t ws_size,
                              hipStream_t stream) {
  const float* input  = (const float*)d_in[0];   // (B,T,1)
  const float* target = (const float*)d_in[1];   // (B,T,1)
  const float* seq    = (const float*)d_in[2];   // (B,T,F)
  const float* fg     = (const float*)d_in[3];   // (B,T,H)

  float* irr = (float*)d_ws;                     // B*T floats  (1 MB)
  float* pP  = irr + (size_t)B * T;              // NB_PEN floats
  float* pM  = pP + NB_PEN;                      // NB_MSE floats

  ewacf_scan_kernel    <<<B * NCHUNK, F,      0, stream>>>(seq, irr);
  ewacf_penalty_kernel <<<NB_PEN,     NT_PEN, 0, stream>>>(fg, irr, pP);
  ewacf_mse_kernel     <<<NB_MSE,     NT_MSE, 0, stream>>>(input, target, pM);
  ewacf_finalize_kernel<<<1,          256,    0, stream>>>(pP, pM, (float*)d_out);

  (void)in_sizes; (void)n_in; (void)out_size; (void)ws_size;
}